// BlockWiseAttention_45818711113838
// MI455X (gfx1250) — compile-verified
//
#include <hip/hip_runtime.h>
#include <math.h>

#define TSEQ 1024
#define NBATCH 4
#define BT (NBATCH * TSEQ)   // 4096 token rows

#define LOG2E 1.44269504088896340736f

typedef float v2f __attribute__((ext_vector_type(2)));
typedef float v8f __attribute__((ext_vector_type(8)));

// V_WMMA_F32_16X16X4_F32 : D(16x16,f32) = A(16x4,f32) * B(4x16,f32) + C
static __device__ inline v8f wmma4(v2f a, v2f b, v8f c) {
  return __builtin_amdgcn_wmma_f32_16x16x4_f32(false, a, false, b, (short)0, c,
                                               false, false);
}

// Raw hardware exp2 (v_exp_f32): 1 TRANS op, no range-check scaffolding.
static __device__ inline float fast_exp2(float x) {
#if __has_builtin(__builtin_amdgcn_exp2f)
  return __builtin_amdgcn_exp2f(x);
#else
  return exp2f(x);
#endif
}

// Combine the two 16-lane halves of the wave (lane i <-> lane i^16).
static __device__ inline float xor16_sum(float v) {
  return v + __shfl_xor(v, 16, 32);
}

static __device__ inline float gelu_exact(float x) {
  return 0.5f * x * (1.f + erff(x * 0.70710678118654752f));
}

// ---------------------------------------------------------------------------
// Kernel A: block-tile gather of M + per-block QKV projection (E=4).
// Q/K/V layout: [block n=16][batch b=4][t=1024][4] (heads packed: cols 2h,2h+1)
// ---------------------------------------------------------------------------
__global__ void blk_qkv_kernel(const float* __restrict__ M,
                               const float* __restrict__ w_in,
                               const float* __restrict__ b_in,
                               float* __restrict__ Q, float* __restrict__ K,
                               float* __restrict__ V) {
  int idx = blockIdx.x * blockDim.x + threadIdx.x;  // [16*4*1024)
  if (idx >= 16 * NBATCH * TSEQ) return;
  int t = idx % TSEQ;
  int b = (idx / TSEQ) % NBATCH;
  int n = idx / (TSEQ * NBATCH);
  int br = n >> 2, bc = n & 3;
  const float* Mrow = M + (size_t)(b * TSEQ + t) * 64;
  float x[4];
#pragma unroll
  for (int i = 0; i < 2; ++i)
#pragma unroll
    for (int j = 0; j < 2; ++j)
      x[2 * i + j] = Mrow[(2 * br + i) * 8 + (2 * bc + j)];
  const float* w = w_in + n * 48;
  const float* bb = b_in + n * 12;
  float o[12];
#pragma unroll
  for (int r = 0; r < 12; ++r) {
    float acc = bb[r];
#pragma unroll
    for (int k = 0; k < 4; ++k) acc += w[r * 4 + k] * x[k];
    o[r] = acc;
  }
#pragma unroll
  for (int e = 0; e < 4; ++e) {
    Q[(size_t)idx * 4 + e] = o[e];
    K[(size_t)idx * 4 + e] = o[4 + e];
    V[(size_t)idx * 4 + e] = o[8 + e];
  }
}

// ---------------------------------------------------------------------------
// Kernel B: per-block attention (hd=2, 2 heads) + out-proj.
// One wave per (block n, batch b, 16-query tile). Computes S^T = K_tile @ Q^T
// via WMMA so the softmax/key axis lands on the in-lane r index; streaming
// exp2-sum with 1/sqrt(hd)*log2(e) folded into Q (logits tiny by construction
// -> shift-free softmax exact in fp32); half-partial accumulators combined
// with a single shfl_xor(16) at the end.
// ---------------------------------------------------------------------------
__global__ __launch_bounds__(32) void blk_attn_kernel(
    const float* __restrict__ Q, const float* __restrict__ K,
    const float* __restrict__ V, const float* __restrict__ w_out,
    const float* __restrict__ b_out, float* __restrict__ all_blocks) {
  int wid = blockIdx.x;
  int qt = wid & 63;
  int b = (wid >> 6) & 3;
  int n = wid >> 8;
  int lane = threadIdx.x;
  int nn = lane & 15;
  int half = lane >> 4;
  int base = (n * NBATCH + b) * TSEQ;
  const float scale = 0.70710678118654752f * LOG2E;  // 1/sqrt(2) * log2(e)

  // B operand = Q^T (lane nn = query column). Head h occupies K-dims 2h,2h+1,
  // i.e. wave-half h; zero the other half to select the head.
  const float4* Q4 = (const float4*)Q;
  float4 qr = Q4[base + qt * 16 + nn];
  v2f qb0, qb1;
  qb0.x = (half == 0) ? qr.x * scale : 0.f;
  qb0.y = (half == 0) ? qr.y * scale : 0.f;
  qb1.x = (half == 1) ? qr.z * scale : 0.f;
  qb1.y = (half == 1) ? qr.w * scale : 0.f;

  const v2f* K2 = (const v2f*)K;    // K row = 2 x v2f
  const float4* V4 = (const float4*)V;

  float l0 = 0.f, l1 = 0.f;                 // half-partial softmax denominators
  float o00 = 0.f, o01 = 0.f, o10 = 0.f, o11 = 0.f;  // half-partial outputs

  for (int kt = 0; kt < 64; ++kt) {
    int krow = base + kt * 16;
    // A operand = K tile (lane m = key index, full 4 dims).
    v2f ka = K2[(size_t)(krow + nn) * 2 + half];
    v8f z = {0.f, 0.f, 0.f, 0.f, 0.f, 0.f, 0.f, 0.f};
    v8f c0 = wmma4(ka, qb0, z);   // S^T head0: c0[r] = s(key kt*16+r+8*half, q nn)
    v8f c1 = wmma4(ka, qb1, z);   // S^T head1
#pragma unroll
    for (int r = 0; r < 8; ++r) {
      float4 vv = V4[krow + r + 8 * half];
      float p0 = fast_exp2(c0[r]);
      float p1 = fast_exp2(c1[r]);
      l0 += p0;
      l1 += p1;
      o00 += p0 * vv.x;
      o01 += p0 * vv.y;
      o10 += p1 * vv.z;
      o11 += p1 * vv.w;
    }
  }
  // combine the two halves (keys 0..7 vs 8..15 of each tile)
  l0 = xor16_sum(l0);
  l1 = xor16_sum(l1);
  float attn0 = xor16_sum(o00) / l0;
  float attn1 = xor16_sum(o01) / l0;
  float attn2 = xor16_sum(o10) / l1;
  float attn3 = xor16_sum(o11) / l1;

  if (half == 0) {  // halves now hold identical values; write once
    float ov[4];
#pragma unroll
    for (int e = 0; e < 4; ++e) {
      const float* we = w_out + n * 16 + e * 4;
      ov[e] = b_out[n * 4 + e] + we[0] * attn0 + we[1] * attn1 +
              we[2] * attn2 + we[3] * attn3;
    }
    int row = b * TSEQ + qt * 16 + nn;
    float4* dst = (float4*)(all_blocks + (size_t)row * 64 + n * 4);
    *dst = make_float4(ov[0], ov[1], ov[2], ov[3]);
  }
}

// ---------------------------------------------------------------------------
// LayerNorm over 64 features, one thread per token row (float4 traffic).
// ---------------------------------------------------------------------------
__global__ void layernorm_kernel(const float* __restrict__ x,
                                 const float* __restrict__ g,
                                 const float* __restrict__ bta,
                                 float* __restrict__ y) {
  int row = blockIdx.x * blockDim.x + threadIdx.x;
  if (row >= BT) return;
  const float4* xr = (const float4*)(x + (size_t)row * 64);
  float4 v[16];
  float mu = 0.f;
#pragma unroll
  for (int i = 0; i < 16; ++i) {
    v[i] = xr[i];
    mu += v[i].x + v[i].y + v[i].z + v[i].w;
  }
  mu *= (1.f / 64.f);
  float var = 0.f;
#pragma unroll
  for (int i = 0; i < 16; ++i) {
    float dx = v[i].x - mu, dy = v[i].y - mu, dz = v[i].z - mu, dw = v[i].w - mu;
    var += dx * dx + dy * dy + dz * dz + dw * dw;
  }
  var *= (1.f / 64.f);
  float inv = rsqrtf(var + 1e-5f);
  float* yr = y + (size_t)row * 64;
#pragma unroll
  for (int i = 0; i < 16; ++i) {
    yr[4 * i + 0] = (v[i].x - mu) * inv * g[4 * i + 0] + bta[4 * i + 0];
    yr[4 * i + 1] = (v[i].y - mu) * inv * g[4 * i + 1] + bta[4 * i + 1];
    yr[4 * i + 2] = (v[i].z - mu) * inv * g[4 * i + 2] + bta[4 * i + 2];
    yr[4 * i + 3] = (v[i].w - mu) * inv * g[4 * i + 3] + bta[4 * i + 3];
  }
}

// ---------------------------------------------------------------------------
// Generic fp32 WMMA GEMM: out[M,N] = epi(A[M,K] @ W[N,K]^T + bias[N]) (+res)
// One wave per 16x16 output tile; K must be a multiple of 4.
// act: 0 = identity, 1 = exact GELU.
// ---------------------------------------------------------------------------
__global__ __launch_bounds__(32) void gemm_wmma_kernel(
    const float* __restrict__ A, const float* __restrict__ W,
    const float* __restrict__ bias, const float* res, float* out, int Ndim,
    int Kdim, int act) {
  int tile = blockIdx.x;
  int ntn = Ndim >> 4;
  int mt = tile / ntn;
  int nt = tile % ntn;
  int lane = threadIdx.x;
  int nn = lane & 15;
  int half = lane >> 4;
  int arow = mt * 16 + nn;  // A-layout: lane m = lane%16
  int col = nt * 16 + nn;   // B/C layout: lane n = lane%16
  v8f c = {0.f, 0.f, 0.f, 0.f, 0.f, 0.f, 0.f, 0.f};
  for (int kc = 0; kc < Kdim; kc += 4) {
    v2f a, bm;
    a.x = A[(size_t)arow * Kdim + kc + 2 * half];
    a.y = A[(size_t)arow * Kdim + kc + 2 * half + 1];
    bm.x = W[(size_t)col * Kdim + kc + 2 * half];
    bm.y = W[(size_t)col * Kdim + kc + 2 * half + 1];
    c = wmma4(a, bm, c);
  }
  float bv = bias[col];
#pragma unroll
  for (int r = 0; r < 8; ++r) {
    int orow = mt * 16 + r + 8 * half;
    float v = c[r] + bv;
    if (act == 1) v = gelu_exact(v);
    if (res) v += res[(size_t)orow * Ndim + col];
    out[(size_t)orow * Ndim + col] = v;
  }
}

// ---------------------------------------------------------------------------
// Kernel D: cross-block attention (E=64, H=4, hd=16).
// One wave per (batch, head, 16-query tile). Per 16-key tile: 4 WMMAs for
// S^T = K@Q^T, single-op exp2 (scale*log2e folded into Q), LDS transpose of
// P, 4 WMMAs for P^T@V accumulated directly (no flash rescale needed).
// ---------------------------------------------------------------------------
__global__ __launch_bounds__(32) void cross_attn_kernel(
    const float* __restrict__ qkv, float* __restrict__ xout) {
  int wid = blockIdx.x;
  int qt = wid & 63;
  int h = (wid >> 6) & 3;
  int b = wid >> 8;
  int lane = threadIdx.x;
  int nn = lane & 15;
  int half = lane >> 4;
  int rbase = b * TSEQ;
  const float scale = 0.25f * LOG2E;  // 1/sqrt(16) * log2(e)

  // B operand (resident): Q^T, lane nn = query column.
  v2f qb[4];
  {
    const float* qp = qkv + (size_t)(rbase + qt * 16 + nn) * 192 + h * 16;
#pragma unroll
    for (int kc = 0; kc < 4; ++kc) {
      qb[kc].x = qp[kc * 4 + 2 * half] * scale;
      qb[kc].y = qp[kc * 4 + 2 * half + 1] * scale;
    }
  }
  __shared__ float Plds[16][17];
  __shared__ float Llds[16];
  float lpart = 0.f;  // half-partial softmax denominator for query nn
  v8f o = {0.f, 0.f, 0.f, 0.f, 0.f, 0.f, 0.f, 0.f};

  for (int kt = 0; kt < 64; ++kt) {
    int k0 = rbase + kt * 16;
    // A operand = K tile (lane m = key index)
    const float* kp = qkv + (size_t)(k0 + nn) * 192 + 64 + h * 16;
    v8f c = {0.f, 0.f, 0.f, 0.f, 0.f, 0.f, 0.f, 0.f};
#pragma unroll
    for (int kc = 0; kc < 4; ++kc) {
      v2f ka;
      ka.x = kp[kc * 4 + 2 * half];
      ka.y = kp[kc * 4 + 2 * half + 1];
      c = wmma4(ka, qb[kc], c);  // S^T[key r+8*half][query nn]
    }
#pragma unroll
    for (int r = 0; r < 8; ++r) {
      float p = fast_exp2(c[r]);
      lpart += p;
      Plds[r + 8 * half][nn] = p;  // Plds[key][query]
    }
    __syncthreads();
    // O[q][d] += P^T @ V : A = P^T chunks from LDS, B = V tile (lane nn = d)
#pragma unroll
    for (int kc = 0; kc < 4; ++kc) {
      v2f pa, vb;
      int kk = kc * 4 + 2 * half;
      pa.x = Plds[kk][nn];
      pa.y = Plds[kk + 1][nn];
      vb.x = qkv[(size_t)(k0 + kk) * 192 + 128 + h * 16 + nn];
      vb.y = qkv[(size_t)(k0 + kk + 1) * 192 + 128 + h * 16 + nn];
      o = wmma4(pa, vb, o);
    }
    __syncthreads();
  }
  float l = xor16_sum(lpart);  // full denominator for query nn (both halves)
  if (half == 0) Llds[nn] = l;
  __syncthreads();
#pragma unroll
  for (int r = 0; r < 8; ++r) {
    int q = r + 8 * half;
    int orow = rbase + qt * 16 + q;
    xout[(size_t)orow * 64 + h * 16 + nn] = o[r] / Llds[q];
  }
}

// ---------------------------------------------------------------------------
// Kernel F: sensitivity gating MLP + final residual blend -> d_out [B,T,8,8]
// ---------------------------------------------------------------------------
__global__ void final_kernel(const float* __restrict__ M,
                             const int* __restrict__ tok,
                             const float* __restrict__ ab,
                             const float* __restrict__ sens_base,
                             const float* __restrict__ sens_emb,
                             const float* __restrict__ w1,
                             const float* __restrict__ b1,
                             const float* __restrict__ w2,
                             const float* __restrict__ b2,
                             float* __restrict__ out) {
  int idx = blockIdx.x * blockDim.x + threadIdx.x;
  if (idx >= BT) return;
  const float* aff = sens_emb + (size_t)tok[idx] * 16;
  float h1[32];
#pragma unroll 4
  for (int r = 0; r < 32; ++r) {
    float acc = b1[r];
#pragma unroll
    for (int k = 0; k < 16; ++k) acc += w1[r * 16 + k] * aff[k];
    h1[r] = gelu_exact(acc);
  }
  float s[16];
#pragma unroll 4
  for (int r = 0; r < 16; ++r) {
    float acc = b2[r];
#pragma unroll
    for (int k = 0; k < 32; ++k) acc += w2[r * 32 + k] * h1[k];
    s[r] = sens_base[r] / (1.f + expf(-acc));
  }
  const float* Mr = M + (size_t)idx * 64;
  const float* abr = ab + (size_t)idx * 64;
  float* outr = out + (size_t)idx * 64;
#pragma unroll 4
  for (int g = 0; g < 16; ++g)
#pragma unroll
    for (int k = 0; k < 4; ++k) {
      int f = 4 * g + k;
      float mv = Mr[f];
      outr[f] = mv + (abr[f] - mv) * s[g];
    }
}

// ---------------------------------------------------------------------------
extern "C" void kernel_launch(void* const* d_in, const int* in_sizes, int n_in,
                              void* d_out, int out_size, void* d_ws,
                              size_t ws_size, hipStream_t stream) {
  (void)in_sizes; (void)n_in; (void)out_size; (void)ws_size;
  const float* M = (const float*)d_in[0];
  const int* tok = (const int*)d_in[1];
  const float* blk_w_in = (const float*)d_in[2];
  const float* blk_b_in = (const float*)d_in[3];
  const float* blk_w_out = (const float*)d_in[4];
  const float* blk_b_out = (const float*)d_in[5];
  const float* x_w_in = (const float*)d_in[6];
  const float* x_b_in = (const float*)d_in[7];
  const float* x_w_out = (const float*)d_in[8];
  const float* x_b_out = (const float*)d_in[9];
  const float* ffn_w1 = (const float*)d_in[10];
  const float* ffn_b1 = (const float*)d_in[11];
  const float* ffn_w2 = (const float*)d_in[12];
  const float* ffn_b2 = (const float*)d_in[13];
  const float* ln1_g = (const float*)d_in[14];
  const float* ln1_b = (const float*)d_in[15];
  const float* ln2_g = (const float*)d_in[16];
  const float* ln2_b = (const float*)d_in[17];
  const float* sens_base = (const float*)d_in[18];
  const float* sens_emb = (const float*)d_in[19];
  const float* sens_w1 = (const float*)d_in[20];
  const float* sens_b1 = (const float*)d_in[21];
  const float* sens_w2 = (const float*)d_in[22];
  const float* sens_b2 = (const float*)d_in[23];

  float* ws = (float*)d_ws;
  float* blkQ = ws;                  // 16*4*1024*4 = 262144
  float* blkK = ws + 262144;         // 262144
  float* blkV = ws + 524288;         // 262144
  float* all_blocks = ws + 786432;   // BT*64 = 262144
  float* lnbuf = ws + 1048576;       // 262144
  float* xqkv = ws + 1310720;        // BT*192 = 786432
  float* xattn = ws + 2097152;       // 262144
  float* ffnh = ws + 2359296;        // BT*256 = 1048576  (end 3407872 floats)

  // 1. per-block QKV projection
  blk_qkv_kernel<<<(16 * NBATCH * TSEQ) / 256, 256, 0, stream>>>(
      M, blk_w_in, blk_b_in, blkQ, blkK, blkV);
  // 2. per-block attention + out-proj -> all_blocks
  blk_attn_kernel<<<16 * NBATCH * 64, 32, 0, stream>>>(
      blkQ, blkK, blkV, blk_w_out, blk_b_out, all_blocks);
  // 3. LN1
  layernorm_kernel<<<BT / 256, 256, 0, stream>>>(all_blocks, ln1_g, ln1_b,
                                                 lnbuf);
  // 4. cross QKV projection: [4096,64] @ [64,192]
  gemm_wmma_kernel<<<(BT / 16) * (192 / 16), 32, 0, stream>>>(
      lnbuf, x_w_in, x_b_in, nullptr, xqkv, 192, 64, 0);
  // 5. cross attention
  cross_attn_kernel<<<NBATCH * 4 * 64, 32, 0, stream>>>(xqkv, xattn);
  // 6. cross out-proj + residual: all_blocks += xattn @ [64,64]
  gemm_wmma_kernel<<<(BT / 16) * (64 / 16), 32, 0, stream>>>(
      xattn, x_w_out, x_b_out, all_blocks, all_blocks, 64, 64, 0);
  // 7. LN2
  layernorm_kernel<<<BT / 256, 256, 0, stream>>>(all_blocks, ln2_g, ln2_b,
                                                 lnbuf);
  // 8. FFN1 + GELU: [4096,64] @ [64,256]
  gemm_wmma_kernel<<<(BT / 16) * (256 / 16), 32, 0, stream>>>(
      lnbuf, ffn_w1, ffn_b1, nullptr, ffnh, 256, 64, 1);
  // 9. FFN2 + residual: [4096,256] @ [256,64]
  gemm_wmma_kernel<<<(BT / 16) * (64 / 16), 32, 0, stream>>>(
      ffnh, ffn_w2, ffn_b2, all_blocks, all_blocks, 64, 256, 0);
  // 10. sensitivity gating + final blend
  final_kernel<<<BT / 256, 256, 0, stream>>>(M, tok, all_blocks, sens_base,
                                             sens_emb, sens_w1, sens_b1,
                                             sens_w2, sens_b2, (float*)d_out);
}